// Stag_74440373174348
// MI455X (gfx1250) — compile-verified
//
#include <hip/hip_runtime.h>
#include <math.h>

#define NN 50000
#define NE 800000
#define IN_F 128
#define HF 128      // H*F
#define H 4
#define F 32
#define EMB 8

typedef float v2f __attribute__((ext_vector_type(2)));
typedef float v8f __attribute__((ext_vector_type(8)));

__device__ __forceinline__ v8f wmma_f32_k4(v2f a, v2f b, v8f c) {
    // D = A(16x4,f32) * B(4x16,f32) + C(16x16,f32)
    return __builtin_amdgcn_wmma_f32_16x16x4_f32(false, a, false, b,
                                                 (short)0, c, false, false);
}

// ---------------------------------------------------------------------------
// Kernel 1: h = feat @ W_fc   [50000,128] x [128,128] -> [50000,128]
// block = 256 thr (8 waves); block handles 16 node rows; wave w -> cols 16w..
// ---------------------------------------------------------------------------
__global__ __launch_bounds__(256) void k_fc(const float* __restrict__ feat,
                                            const float* __restrict__ Wfc,
                                            float* __restrict__ hbuf) {
    const int lane = threadIdx.x & 31;
    const int wave = threadIdx.x >> 5;
    const int row0 = blockIdx.x << 4;
    const int col0 = wave << 4;
    const int m    = lane & 15;
    const int kh   = lane >> 4;          // K-half select (0/1)
    const float* arow = feat + (size_t)(row0 + m) * IN_F;
    v8f c = {};
#pragma unroll
    for (int k0 = 0; k0 < IN_F; k0 += 4) {
        v2f a = *(const v2f*)(arow + k0 + 2 * kh);          // K = 2kh, 2kh+1
        v2f b;
        b.x = Wfc[(size_t)(k0 + 2 * kh) * HF + col0 + m];
        b.y = Wfc[(size_t)(k0 + 2 * kh + 1) * HF + col0 + m];
        c = wmma_f32_k4(a, b, c);
    }
#pragma unroll
    for (int v = 0; v < 8; ++v)
        hbuf[(size_t)(row0 + v + 8 * kh) * HF + col0 + m] = c[v];
}

// ---------------------------------------------------------------------------
// Kernel 2: per-head post = h[:,hd,:] @ W_post + b_post  ([N,32]x[32,32])
// wave w: hd = w>>1, col-tile ct = w&1. ct==0 -> loc_l/loc_r (cols 0..15),
// ct==1 -> lsl/lsr (cols 16..31), reduced in-wave to S_l/S_r scalars.
// ---------------------------------------------------------------------------
__global__ __launch_bounds__(256) void k_post(const float* __restrict__ hbuf,
                                              const float* __restrict__ Wpost,
                                              const float* __restrict__ bpost,
                                              float* __restrict__ locl,
                                              float* __restrict__ locr,
                                              float* __restrict__ Sl,
                                              float* __restrict__ Sr) {
    const int lane = threadIdx.x & 31;
    const int wave = threadIdx.x >> 5;
    const int hd   = wave >> 1;
    const int ct   = wave & 1;
    const int row0 = blockIdx.x << 4;
    const int m    = lane & 15;
    const int kh   = lane >> 4;
    const float* arow = hbuf + (size_t)(row0 + m) * HF + hd * F;
    v8f c = {};
#pragma unroll
    for (int k0 = 0; k0 < F; k0 += 4) {
        v2f a = *(const v2f*)(arow + k0 + 2 * kh);
        v2f b;
        b.x = Wpost[(size_t)(k0 + 2 * kh) * (4 * EMB) + ct * 16 + m];
        b.y = Wpost[(size_t)(k0 + 2 * kh + 1) * (4 * EMB) + ct * 16 + m];
        c = wmma_f32_k4(a, b, c);
    }
    const float bc = bpost[ct * 16 + m];
    if (ct == 0) {
        // cols 0..7 -> loc_l, cols 8..15 -> loc_r
#pragma unroll
        for (int v = 0; v < 8; ++v) {
            float val = c[v] + bc;
            int row = row0 + v + 8 * kh;
            if (m < 8) locl[(size_t)row * (H * EMB) + hd * EMB + m] = val;
            else       locr[(size_t)row * (H * EMB) + hd * EMB + (m - 8)] = val;
        }
    } else {
        // cols 16..23 -> sum -> S_l ; cols 24..31 -> sum -> S_r
#pragma unroll
        for (int v = 0; v < 8; ++v) {
            float s = c[v] + bc;
            s += __shfl_xor(s, 1);
            s += __shfl_xor(s, 2);
            s += __shfl_xor(s, 4);       // groups of 8 lanes now hold sum
            if ((lane & 7) == 0) {
                int row = row0 + v + 8 * kh;
                if (((lane >> 3) & 1) == 0) Sl[(size_t)row * H + hd] = s;
                else                        Sr[(size_t)row * H + hd] = s;
            }
        }
    }
}

// ---------------------------------------------------------------------------
// Kernel 3: per (edge,head): e = dot8(loc_l[src],loc_r[dst]) +
//           exp(S_l[src]+S_r[dst]) * eps ; segment-max into mbuf via atomics
// ---------------------------------------------------------------------------
__device__ __forceinline__ void atomicMaxF32(float* addr, float v) {
    if (v >= 0.0f) atomicMax((int*)addr, __float_as_int(v));
    else atomicMin((unsigned int*)addr, (unsigned int)__float_as_int(v));
}

__global__ __launch_bounds__(256) void k_edge(const int* __restrict__ src,
                                              const int* __restrict__ dst,
                                              const float* __restrict__ locl,
                                              const float* __restrict__ locr,
                                              const float* __restrict__ Sl,
                                              const float* __restrict__ Sr,
                                              const float* __restrict__ eps,
                                              float* __restrict__ ebuf,
                                              float* __restrict__ mbuf) {
    int i = blockIdx.x * 256 + threadIdx.x;
    if (i >= NE * H) return;
    int e = i >> 2, hd = i & 3;
    int s = src[e], d = dst[e];
    const float4 l0 = *(const float4*)(locl + (size_t)s * (H * EMB) + hd * EMB);
    const float4 l1 = *(const float4*)(locl + (size_t)s * (H * EMB) + hd * EMB + 4);
    const float4 r0 = *(const float4*)(locr + (size_t)d * (H * EMB) + hd * EMB);
    const float4 r1 = *(const float4*)(locr + (size_t)d * (H * EMB) + hd * EMB + 4);
    float loc = l0.x * r0.x + l0.y * r0.y + l0.z * r0.z + l0.w * r0.w +
                l1.x * r1.x + l1.y * r1.y + l1.z * r1.z + l1.w * r1.w;
    float scale = __expf(Sl[s * H + hd] + Sr[d * H + hd]);
    float ev = loc + scale * eps[i];
    ebuf[i] = ev;
    atomicMaxF32(mbuf + (size_t)d * H + hd, ev);
}

// ---------------------------------------------------------------------------
// Kernel 4: ee = exp(e - m[dst]); segment-sum into denom
// ---------------------------------------------------------------------------
__global__ __launch_bounds__(256) void k_norm(const int* __restrict__ dst,
                                              const float* __restrict__ mbuf,
                                              float* __restrict__ ebuf,
                                              float* __restrict__ denom) {
    int i = blockIdx.x * 256 + threadIdx.x;
    if (i >= NE * H) return;
    int e = i >> 2, hd = i & 3;
    int d = dst[e];
    float ee = __expf(ebuf[i] - mbuf[(size_t)d * H + hd]);
    ebuf[i] = ee;
    atomicAdd(denom + (size_t)d * H + hd, ee);
}

// ---------------------------------------------------------------------------
// Kernel 5: out[dst] += h[src] * (ee/denom[dst]); out pre-seeded with bias.
// h and out both L2-resident (25.6 MB each, 192 MB L2).
// ---------------------------------------------------------------------------
__global__ __launch_bounds__(256) void k_scatter(const int* __restrict__ src,
                                                 const int* __restrict__ dst,
                                                 const float* __restrict__ hbuf,
                                                 const float* __restrict__ ebuf,
                                                 const float* __restrict__ denom,
                                                 float* __restrict__ out) {
    int i = blockIdx.x * 256 + threadIdx.x;   // over NE*128 = 102.4M
    int f  = i & 31;
    int hd = (i >> 5) & 3;
    int e  = i >> 7;
    if (e >= NE) return;
    int s = src[e], d = dst[e];
    float a = ebuf[(size_t)e * H + hd] / denom[(size_t)d * H + hd];
    float val = hbuf[(size_t)s * HF + hd * F + f] * a;
    atomicAdd(out + (size_t)d * HF + hd * F + f, val);
}

// ---------------------------------------------------------------------------
// Kernel 0: out = bias (broadcast), m = -inf, denom = 0
// ---------------------------------------------------------------------------
__global__ __launch_bounds__(256) void k_init(const float* __restrict__ bias,
                                              float* __restrict__ out,
                                              float* __restrict__ mbuf,
                                              float* __restrict__ denom) {
    int i = blockIdx.x * 256 + threadIdx.x;
    if (i < NN * HF) out[i] = bias[i & 127];
    if (i < NN * H) { mbuf[i] = -__builtin_inff(); denom[i] = 0.0f; }
}

extern "C" void kernel_launch(void* const* d_in, const int* in_sizes, int n_in,
                              void* d_out, int out_size, void* d_ws, size_t ws_size,
                              hipStream_t stream) {
    const float* feat  = (const float*)d_in[0];
    const float* Wfc   = (const float*)d_in[1];
    const float* Wpost = (const float*)d_in[2];
    const float* bpost = (const float*)d_in[3];
    const float* bias  = (const float*)d_in[4];
    const float* eps   = (const float*)d_in[5];
    const int*   src   = (const int*)d_in[6];
    const int*   dst   = (const int*)d_in[7];
    float* out = (float*)d_out;

    float* ws = (float*)d_ws;
    size_t o = 0;
    float* hbuf  = ws + o; o += (size_t)NN * HF;        // 6.4M
    float* locl  = ws + o; o += (size_t)NN * H * EMB;   // 1.6M
    float* locr  = ws + o; o += (size_t)NN * H * EMB;   // 1.6M
    float* Sl    = ws + o; o += (size_t)NN * H;         // 0.2M
    float* Sr    = ws + o; o += (size_t)NN * H;         // 0.2M
    float* ebuf  = ws + o; o += (size_t)NE * H;         // 3.2M
    float* mbuf  = ws + o; o += (size_t)NN * H;         // 0.2M
    float* denom = ws + o;                              // 0.2M  (~54 MB total)

    k_init   <<<(NN * HF + 255) / 256, 256, 0, stream>>>(bias, out, mbuf, denom);
    k_fc     <<<NN / 16,               256, 0, stream>>>(feat, Wfc, hbuf);
    k_post   <<<NN / 16,               256, 0, stream>>>(hbuf, Wpost, bpost,
                                                         locl, locr, Sl, Sr);
    k_edge   <<<(NE * H + 255) / 256,  256, 0, stream>>>(src, dst, locl, locr,
                                                         Sl, Sr, eps, ebuf, mbuf);
    k_norm   <<<(NE * H + 255) / 256,  256, 0, stream>>>(dst, mbuf, ebuf, denom);
    k_scatter<<<(NE * HF + 255) / 256, 256, 0, stream>>>(src, dst, hbuf, ebuf,
                                                         denom, out);
}